// denseconv_16501264351992
// MI455X (gfx1250) — compile-verified
//
#include <hip/hip_runtime.h>
#include <hip/hip_bf16.h>

typedef __attribute__((ext_vector_type(2))) float v2f;
typedef __attribute__((ext_vector_type(4))) float v4f;
typedef __attribute__((ext_vector_type(8))) float v8f;

#define BATCH 4
#define DIMS  64
#define NPTS  4096
#define KNN   16
#define GRP   64
#define XCOL  388          // column stride in dwords: 388 % 64 == 4 (4*odd) ->
                           // banks (4*col + row%4) are a bijection: conflict-free

// X layout per point (column-major, 16 columns of XCOL dwords):
//   rows   0.. 63 : H3    rows  64..127 : H2    rows 128..191 : H1
//   rows 192..319 : X0 (central | nbr-central)   rows 320..383 : Z1 scratch
#define XI(row, col) ((col) * XCOL + (row))

// ---------------------------------------------------------------------------
// Kernel 0: per-point squared norms  norms[b*N+n] = sum_d x[b][d][n]^2
// ---------------------------------------------------------------------------
__global__ __launch_bounds__(256) void norms_kernel(const float* __restrict__ x,
                                                    float* __restrict__ norms) {
    int p = blockIdx.x * 256 + threadIdx.x;           // 0..16383
    int b = p >> 12, n = p & (NPTS - 1);
    float s = 0.f;
    #pragma unroll 8
    for (int d = 0; d < DIMS; ++d) {
        float v = x[(b * DIMS + d) * NPTS + n];
        s = fmaf(v, v, s);
    }
    norms[p] = s;
}

// ---------------------------------------------------------------------------
// Kernel 1: KNN — one wave per query point. Candidate tiles staged in LDS and
// shared by the 4 waves in the block. Per-lane sorted top-17 insertion list,
// then a 17-round wave-wide min merge (shuffle reduction with owner tracking).
// Writes idx (B,K,N) to workspace and to the int32 tail of d_out.
// ---------------------------------------------------------------------------
__global__ __launch_bounds__(128) void knn_kernel(const float* __restrict__ x,
                                                  const float* __restrict__ norms,
                                                  int* __restrict__ idx_ws,
                                                  int* __restrict__ idx_out) {
    __shared__ float cand[64 * 64];          // [d][j] tile of 64 candidates
    __shared__ float qv[4][64];              // per-wave query vector
    __shared__ float md[4][32 * 17];         // per-lane merge lists (dist)
    __shared__ int   mi[4][32 * 17];         // per-lane merge lists (index)

    const int tid  = threadIdx.x;
    const int wave = tid >> 5;
    const int lane = tid & 31;
    const int point = blockIdx.x * 4 + wave;          // 0..16383
    const int b = point >> 12, n = point & (NPTS - 1);

    qv[wave][lane]      = x[(b * DIMS + lane) * NPTS + n];
    qv[wave][lane + 32] = x[(b * DIMS + lane + 32) * NPTS + n];
    const float qn = norms[point];

    float dist[17];
    int   ind[17];
    #pragma unroll
    for (int p = 0; p < 17; ++p) { dist[p] = 3.4e38f; ind[p] = 0; }

    for (int base = 0; base < NPTS; base += 64) {
        __syncthreads();
        // cooperative coalesced tile load: 64 dims x 64 candidates
        #pragma unroll
        for (int e = tid; e < 4096; e += 128) {
            int d = e >> 6, j = e & 63;
            cand[(d << 6) | j] = x[(b * DIMS + d) * NPTS + base + j];
        }
        __syncthreads();
        #pragma unroll
        for (int s = 0; s < 2; ++s) {
            const int j = lane + s * 32;
            float dot = 0.f;
            #pragma unroll
            for (int d = 0; d < DIMS; ++d)
                dot = fmaf(qv[wave][d], cand[(d << 6) | j], dot);
            const float rn = norms[b * NPTS + base + j];
            const float d2 = qn + rn - 2.f * dot;
            const int ci = base + j;
            if (d2 < dist[16]) {                       // insertion into sorted 17-list
                dist[16] = d2; ind[16] = ci;
                #pragma unroll
                for (int p = 16; p > 0; --p) {
                    if (dist[p] < dist[p - 1]) {
                        float td = dist[p]; dist[p] = dist[p - 1]; dist[p - 1] = td;
                        int   ti = ind[p];  ind[p]  = ind[p - 1];  ind[p - 1]  = ti;
                    }
                }
            }
        }
    }

    // spill per-lane lists to LDS (own slots only; same-wave DS ops are in order)
    #pragma unroll
    for (int p = 0; p < 17; ++p) {
        md[wave][lane * 17 + p] = dist[p];
        mi[wave][lane * 17 + p] = ind[p];
    }

    int p = 0;
    for (int t = 0; t < 17; ++t) {
        float h  = md[wave][lane * 17 + p];
        int   hi = mi[wave][lane * 17 + p];
        int   owner = lane;
        #pragma unroll
        for (int off = 16; off; off >>= 1) {
            float oh  = __shfl_xor(h, off, 32);
            int   ohi = __shfl_xor(hi, off, 32);
            int   oo  = __shfl_xor(owner, off, 32);
            if (oh < h || (oh == h && oo < owner)) { h = oh; hi = ohi; owner = oo; }
        }
        if (lane == 0 && t > 0) {                      // drop t==0 (self)
            idx_ws [(b * KNN + (t - 1)) * NPTS + n] = hi;
            idx_out[(b * KNN + (t - 1)) * NPTS + n] = hi;
        }
        if (lane == owner) p++;
    }
}

// ---------------------------------------------------------------------------
// Kernel 2: fused edge-feature + conv chain. One wave owns TWO adjacent
// points; each point's K=16 edge columns form one WMMA N-tile. A (weight)
// fragments load once and feed two WMMAs; B fragments are single ds_load_b64
// (column-major X, conflict-free 388-dword column stride); C fragments are
// written as two ds_store_b128 per tile.
// ---------------------------------------------------------------------------
__device__ __forceinline__ v8f wmma_f32x4(v2f a, v2f b, v8f c) {
    return __builtin_amdgcn_wmma_f32_16x16x4_f32(false, a, false, b, (short)0, c,
                                                 false, false);
}

__device__ __forceinline__ v2f ldsv2(const float* p) { return *(const v2f*)p; }

__device__ __forceinline__ void stv8(float* dst, v8f a, const float* bias, int rowbase,
                                     bool relu) {
    v4f lo, hi;
    #pragma unroll
    for (int r = 0; r < 4; ++r) {
        float vl = a[r]     + bias[rowbase + r];
        float vh = a[r + 4] + bias[rowbase + r + 4];
        lo[r] = relu ? fmaxf(vl, 0.f) : vl;
        hi[r] = relu ? fmaxf(vh, 0.f) : vh;
    }
    *(v4f*)(dst)     = lo;                              // ds_store_b128
    *(v4f*)(dst + 4) = hi;                              // ds_store_b128
}

__device__ __forceinline__ void conv_layer2(float* __restrict__ Xa,
                                            float* __restrict__ Xb,
                                            const float* __restrict__ W,
                                            const float* __restrict__ bias,
                                            int inbase, int inlen, int outbase,
                                            int col, int half) {
    #pragma unroll
    for (int mt = 0; mt < 4; ++mt) {                   // 64 output rows = 4 M-tiles
        v8f acc0 = {}, acc1 = {};
        for (int kk = 0; kk < inlen; kk += 4) {
            v2f A  = { W[(mt * 16 + col) * inlen + kk + 2 * half],
                       W[(mt * 16 + col) * inlen + kk + 1 + 2 * half] };
            v2f B0 = ldsv2(Xa + XI(inbase + kk + 2 * half, col));   // ds_load_b64
            v2f B1 = ldsv2(Xb + XI(inbase + kk + 2 * half, col));
            acc0 = wmma_f32x4(A, B0, acc0);
            acc1 = wmma_f32x4(A, B1, acc1);
        }
        const int rb = mt * 16 + 8 * half;             // C/D rows rb..rb+7
        stv8(Xa + XI(outbase + rb, col), acc0, bias, rb, true);
        stv8(Xb + XI(outbase + rb, col), acc1, bias, rb, true);
    }
    __syncthreads();
}

__global__ __launch_bounds__(32) void fused_kernel(
    const float* __restrict__ x, const int* __restrict__ idxw,
    const float* __restrict__ w1, const float* __restrict__ b1,
    const float* __restrict__ w2, const float* __restrict__ b2,
    const float* __restrict__ w3, const float* __restrict__ b3,
    const float* __restrict__ w5a, const float* __restrict__ b5a,
    const float* __restrict__ w5b, const float* __restrict__ b5b,
    float* __restrict__ out) {
    __shared__ float Xs[2][16 * XCOL];                 // 49,664 B: two points

    const int lane = threadIdx.x & 31;
    const int p0 = blockIdx.x * 2;                     // two adjacent points
    const int b  = p0 >> 12;
    const int n0 = p0 & (NPTS - 1);
    const int n1 = n0 + 1;
    float* Xa = &Xs[0][0];
    float* Xb = &Xs[1][0];
    const int col = lane & 15, half = lane >> 4;

    // warm L2 with the weight panels (global_prefetch_b8)
    __builtin_prefetch(w1  + lane * 256, 0, 3);        // 8192  floats
    __builtin_prefetch(w2  + lane * 384, 0, 3);        // 12288 floats
    __builtin_prefetch(w3  + lane * 512, 0, 3);        // 16384 floats
    __builtin_prefetch(w5a + lane * 128, 0, 3);        // 4096  floats
    __builtin_prefetch(w5b + lane * 128, 0, 3);        // 4096  floats

    // --- build X0: rows 192..255 central, 256..319 neighbor - central -------
    const int nbr0 = idxw[(b * KNN + col) * NPTS + n0];
    const int nbr1 = idxw[(b * KNN + col) * NPTS + n1];
    for (int dd = 0; dd < 32; dd += 4) {
        const int d = dd + 32 * half;
        v4f c0, c1, e0, e1;
        #pragma unroll
        for (int r = 0; r < 4; ++r) {
            const float* xr = x + (b * DIMS + d + r) * NPTS;
            float a0 = xr[n0], a1 = xr[n1];
            c0[r] = a0;             c1[r] = a1;
            e0[r] = xr[nbr0] - a0;  e1[r] = xr[nbr1] - a1;
        }
        *(v4f*)(Xa + XI(192 + d, col)) = c0;           // ds_store_b128
        *(v4f*)(Xb + XI(192 + d, col)) = c1;
        *(v4f*)(Xa + XI(256 + d, col)) = e0;
        *(v4f*)(Xb + XI(256 + d, col)) = e1;
    }
    __syncthreads();

    // --- dense conv chain (WMMA f32 16x16x4, A shared across both points) ---
    conv_layer2(Xa, Xb, w1, b1, /*in*/192, 128, /*out*/128, col, half);   // H1
    conv_layer2(Xa, Xb, w2, b2, /*in*/128, 192, /*out*/ 64, col, half);   // H2
    conv_layer2(Xa, Xb, w3, b3, /*in*/ 64, 256, /*out*/  0, col, half);   // H3

    // --- grouped 64x64 convs per m, fused max over K ------------------------
    for (int m = 0; m < 5; ++m) {
        // Z1 = relu(w5a * X3[:, m-slice] + b5a)  -> rows 320..383
        #pragma unroll
        for (int mt = 0; mt < 4; ++mt) {
            v8f acc0 = {}, acc1 = {};
            for (int kk = 0; kk < 64; kk += 4) {
                v2f A  = { w5a[(mt * 16 + col) * 64 + kk + 2 * half],
                           w5a[(mt * 16 + col) * 64 + kk + 1 + 2 * half] };
                // m-slice rows differ by 5: two b32 reads (ds_load_2addr-able)
                v2f B0 = { Xa[XI(5 * (kk + 2 * half) + m, col)],
                           Xa[XI(5 * (kk + 1 + 2 * half) + m, col)] };
                v2f B1 = { Xb[XI(5 * (kk + 2 * half) + m, col)],
                           Xb[XI(5 * (kk + 1 + 2 * half) + m, col)] };
                acc0 = wmma_f32x4(A, B0, acc0);
                acc1 = wmma_f32x4(A, B1, acc1);
            }
            const int rb = mt * 16 + 8 * half;
            stv8(Xa + XI(320 + rb, col), acc0, b5a, rb, true);
            stv8(Xb + XI(320 + rb, col), acc1, b5a, rb, true);
        }
        __syncthreads();
        // Z2 = w5b * Z1 + b5b, then max over the 16 k-columns, store
        #pragma unroll
        for (int mt = 0; mt < 4; ++mt) {
            v8f acc0 = {}, acc1 = {};
            for (int kk = 0; kk < 64; kk += 4) {
                v2f A  = { w5b[(mt * 16 + col) * 64 + kk + 2 * half],
                           w5b[(mt * 16 + col) * 64 + kk + 1 + 2 * half] };
                v2f B0 = ldsv2(Xa + XI(320 + kk + 2 * half, col));   // ds_load_b64
                v2f B1 = ldsv2(Xb + XI(320 + kk + 2 * half, col));
                acc0 = wmma_f32x4(A, B0, acc0);
                acc1 = wmma_f32x4(A, B1, acc1);
            }
            #pragma unroll
            for (int r = 0; r < 8; ++r) {
                int o = mt * 16 + r + 8 * half;
                float bv = b5b[o];
                float v0 = acc0[r] + bv;
                float v1 = acc1[r] + bv;
                // max over columns (k) within the 16-lane half
                #pragma unroll
                for (int off = 1; off < 16; off <<= 1) {
                    v0 = fmaxf(v0, __shfl_xor(v0, off, 32));
                    v1 = fmaxf(v1, __shfl_xor(v1, off, 32));
                }
                if (col == 0) {
                    out[(b * 320 + o * 5 + m) * NPTS + n0] = v0;
                    out[(b * 320 + o * 5 + m) * NPTS + n1] = v1;
                }
            }
        }
        __syncthreads();
    }
}

// ---------------------------------------------------------------------------
extern "C" void kernel_launch(void* const* d_in, const int* in_sizes, int n_in,
                              void* d_out, int out_size, void* d_ws, size_t ws_size,
                              hipStream_t stream) {
    const float* x   = (const float*)d_in[0];
    const float* w1  = (const float*)d_in[1];
    const float* b1  = (const float*)d_in[2];
    const float* w2  = (const float*)d_in[3];
    const float* b2  = (const float*)d_in[4];
    const float* w3  = (const float*)d_in[5];
    const float* b3  = (const float*)d_in[6];
    const float* w5a = (const float*)d_in[7];
    const float* b5a = (const float*)d_in[8];
    const float* w5b = (const float*)d_in[9];
    const float* b5b = (const float*)d_in[10];

    float* out    = (float*)d_out;
    float* norms  = (float*)d_ws;                              // 16384 floats
    int*   idxw   = (int*)((char*)d_ws + BATCH * NPTS * 4);    // B*K*N ints
    int*   idxout = (int*)(out + BATCH * 320 * NPTS);          // output tail

    norms_kernel<<<dim3((BATCH * NPTS) / 256), dim3(256), 0, stream>>>(x, norms);
    knn_kernel<<<dim3((BATCH * NPTS) / 4), dim3(128), 0, stream>>>(x, norms, idxw, idxout);
    fused_kernel<<<dim3((BATCH * NPTS) / 2), dim3(32), 0, stream>>>(
        x, idxw, w1, b1, w2, b2, w3, b3, w5a, b5a, w5b, b5b, out);
    (void)in_sizes; (void)n_in; (void)out_size; (void)ws_size;
}